// MLPPredictor_30451318129156
// MI455X (gfx1250) — compile-verified
//
#include <hip/hip_runtime.h>
#include <hip/hip_bf16.h>

typedef __attribute__((ext_vector_type(2))) float v2f;
typedef __attribute__((ext_vector_type(8))) float v8f;

// -----------------------------------------------------------------------------
// Kernel 1: node projection GEMM with f32 WMMA.
//   P[n, 0:128]   = h[n,:] @ W1[:, 0:128]^T    (contribution when n is src)
//   P[n, 128:256] = h[n,:] @ W1[:, 128:256]^T  (contribution when n is dst)
// W1 is [128, 256] row-major (torch Linear weight). Each block: 16 nodes,
// 8 waves; wave w computes output columns [32w, 32w+32) as two 16x16 tiles,
// accumulating over K=128 in steps of 4 (V_WMMA_F32_16X16X4_F32).
// -----------------------------------------------------------------------------
__global__ __launch_bounds__(256) void node_proj_wmma(
    const float* __restrict__ h,    // [n_nodes, 128]
    const float* __restrict__ W1,   // [128, 256]
    float* __restrict__ P,          // [n_nodes, 256]
    int n_nodes)
{
    const int node0 = blockIdx.x << 4;          // 16 nodes per block
    const int wave  = threadIdx.x >> 5;         // 0..7
    const int lane  = threadIdx.x & 31;
    const int l     = lane & 15;
    const int kk    = (lane >> 4) << 1;         // 0 (lanes 0-15) or 2 (lanes 16-31)

    const int o0    = wave << 5;                // output col base: 0,32,...,224
    const int half  = o0 >> 7;                  // 0 -> src-half, 1 -> dst-half
    const int koff  = half << 7;                // column offset into W1's K dim
    const int orow0 = o0 & 127;                 // W1 output-row for tile 0
    const int orow1 = (o0 + 16) & 127;          // W1 output-row for tile 1

    int hrow = node0 + l; if (hrow >= n_nodes) hrow = n_nodes - 1;
    const float* hp  = h  + (size_t)hrow * 128 + kk;
    const float* wp0 = W1 + (size_t)(orow0 + l) * 256 + koff + kk;
    const float* wp1 = W1 + (size_t)(orow1 + l) * 256 + koff + kk;

    v8f c0 = {}; v8f c1 = {};
    #pragma unroll 4
    for (int kb = 0; kb < 128; kb += 4) {
        // A fragment: 16x4, lane (l, half) holds A[M=l, K=kb+kk .. kb+kk+1]
        v2f a;  a.x  = hp[kb];  a.y  = hp[kb + 1];
        // B fragments: 4x16, lane (l, half) holds B[K=kb+kk.., N=l]
        v2f b0; b0.x = wp0[kb]; b0.y = wp0[kb + 1];
        v2f b1; b1.x = wp1[kb]; b1.y = wp1[kb + 1];
        c0 = __builtin_amdgcn_wmma_f32_16x16x4_f32(false, a, false, b0,
                                                   (short)0, c0, false, false);
        c1 = __builtin_amdgcn_wmma_f32_16x16x4_f32(false, a, false, b1,
                                                   (short)0, c1, false, false);
    }

    // D layout: VGPR r -> rows (r, r+8) for lanes (0-15, 16-31), N = lane&15.
    // Uniform tile-level guard: the ragged path only runs for a partial last
    // tile (never for n_nodes % 16 == 0), keeping the hot path branch-free.
    const int mbase = (lane >> 4) << 3;
    if (node0 + 16 <= n_nodes) {
        #pragma unroll
        for (int r = 0; r < 8; ++r) {
            float* Pr = P + (size_t)(node0 + r + mbase) * 256;
            Pr[o0 + l]      = c0[r];
            Pr[o0 + 16 + l] = c1[r];
        }
    } else {
        #pragma unroll
        for (int r = 0; r < 8; ++r) {
            const int m = node0 + r + mbase;
            if (m < n_nodes) {
                float* Pr = P + (size_t)m * 256;
                Pr[o0 + l]      = c0[r];
                Pr[o0 + 16 + l] = c1[r];
            }
        }
    }
}

// -----------------------------------------------------------------------------
// Kernel 2: per-edge fused relu-dot. One wave32 per TWO edges per iteration;
// lane l handles feature dims [4l, 4l+4) with float4 loads (coalesced 512B
// per P row). Four independent P-row loads in flight per iteration for L2
// latency hiding; the two shfl_xor reduction chains are interleaved and both
// scores are written by a single 2-lane store.
// -----------------------------------------------------------------------------
__device__ __forceinline__ float relu_dot4(const float4 x1, const float4 x2,
                                           const float4 bb, const float4 w2) {
    return w2.x * fmaxf(x1.x + x2.x + bb.x, 0.0f)
         + w2.y * fmaxf(x1.y + x2.y + bb.y, 0.0f)
         + w2.z * fmaxf(x1.z + x2.z + bb.z, 0.0f)
         + w2.w * fmaxf(x1.w + x2.w + bb.w, 0.0f);
}

__global__ __launch_bounds__(256) void edge_score_kernel(
    const long long* __restrict__ src,
    const long long* __restrict__ dst,
    const float* __restrict__ P,     // [n_nodes, 256]
    const float* __restrict__ b1,    // [128]
    const float* __restrict__ W2,    // [128]
    const float* __restrict__ b2,    // [1]
    float* __restrict__ out,         // [n_edges]
    int n_edges)
{
    const int lane   = threadIdx.x & 31;
    const int waveId = blockIdx.x * (blockDim.x >> 5) + (threadIdx.x >> 5);
    const int nWaves = gridDim.x * (blockDim.x >> 5);

    const float4 w2 = ((const float4*)W2)[lane];   // W2[4l..4l+3]
    const float4 bb = ((const float4*)b1)[lane];   // b1[4l..4l+3]
    const float beta = b2[0];

    for (int e = waveId * 2; e < n_edges; e += nWaves * 2) {
        if (e + 1 < n_edges) {
            // e is even and stride is even -> 16B-aligned paired index loads
            const longlong2 sp = ((const longlong2*)src)[e >> 1];
            const longlong2 dp = ((const longlong2*)dst)[e >> 1];
            const float4 a0 = ((const float4*)(P + (size_t)sp.x * 256))[lane];
            const float4 a1 = ((const float4*)(P + (size_t)sp.y * 256))[lane];
            const float4 c0 = ((const float4*)(P + (size_t)dp.x * 256 + 128))[lane];
            const float4 c1 = ((const float4*)(P + (size_t)dp.y * 256 + 128))[lane];

            float acc0 = relu_dot4(a0, c0, bb, w2);
            float acc1 = relu_dot4(a1, c1, bb, w2);
            #pragma unroll
            for (int off = 16; off > 0; off >>= 1) {
                acc0 += __shfl_xor(acc0, off, 32);
                acc1 += __shfl_xor(acc1, off, 32);
            }
            if (lane < 2) out[e + lane] = (lane ? acc1 : acc0) + beta;
        } else {
            const size_t s = (size_t)src[e];
            const size_t d = (size_t)dst[e];
            const float4 x1 = ((const float4*)(P + s * 256))[lane];
            const float4 x2 = ((const float4*)(P + d * 256 + 128))[lane];
            float acc = relu_dot4(x1, x2, bb, w2);
            #pragma unroll
            for (int off = 16; off > 0; off >>= 1)
                acc += __shfl_xor(acc, off, 32);
            if (lane == 0) out[e] = acc + beta;
        }
    }
}

// -----------------------------------------------------------------------------
// Fallback (only if the workspace is too small for P): direct per-edge MLP,
// one thread per edge. Deterministic; never used when ws is adequate.
// -----------------------------------------------------------------------------
__global__ __launch_bounds__(256) void edge_direct_kernel(
    const float* __restrict__ h,
    const long long* __restrict__ src,
    const long long* __restrict__ dst,
    const float* __restrict__ W1,   // [128, 256]
    const float* __restrict__ b1,   // [128]
    const float* __restrict__ W2,   // [128]
    const float* __restrict__ b2,   // [1]
    float* __restrict__ out,
    int n_edges)
{
    const int stride = gridDim.x * blockDim.x;
    for (int e = blockIdx.x * blockDim.x + threadIdx.x; e < n_edges; e += stride) {
        const size_t s = (size_t)src[e];
        const size_t d = (size_t)dst[e];
        const float4* hs = (const float4*)(h + s * 128);
        const float4* hd = (const float4*)(h + d * 128);
        float acc = b2[0];
        for (int o = 0; o < 128; ++o) {
            const float4* w = (const float4*)(W1 + (size_t)o * 256);
            float t = b1[o];
            #pragma unroll 4
            for (int k = 0; k < 32; ++k) {
                const float4 a = hs[k], bw = w[k];
                t += a.x * bw.x + a.y * bw.y + a.z * bw.z + a.w * bw.w;
                const float4 a2 = hd[k], bw2 = w[32 + k];
                t += a2.x * bw2.x + a2.y * bw2.y + a2.z * bw2.z + a2.w * bw2.w;
            }
            acc += W2[o] * fmaxf(t, 0.0f);
        }
        out[e] = acc;
    }
}

extern "C" void kernel_launch(void* const* d_in, const int* in_sizes, int n_in,
                              void* d_out, int out_size, void* d_ws, size_t ws_size,
                              hipStream_t stream) {
    const float*     h   = (const float*)d_in[0];      // [N, 128] f32
    const long long* src = (const long long*)d_in[1];  // [E] i64
    const long long* dst = (const long long*)d_in[2];  // [E] i64
    const float*     W1  = (const float*)d_in[3];      // [128, 256] f32
    const float*     b1  = (const float*)d_in[4];      // [128] f32
    const float*     W2  = (const float*)d_in[5];      // [128] f32
    const float*     b2  = (const float*)d_in[6];      // [1] f32
    float*           out = (float*)d_out;

    const int n_nodes = in_sizes[0] / 128;
    const int n_edges = in_sizes[1];

    const size_t need = (size_t)n_nodes * 256 * sizeof(float);
    if (d_ws != nullptr && ws_size >= need) {
        float* P = (float*)d_ws;
        const int nblocks = (n_nodes + 15) / 16;
        node_proj_wmma<<<nblocks, 256, 0, stream>>>(h, W1, P, n_nodes);
        edge_score_kernel<<<2048, 256, 0, stream>>>(src, dst, P, b1, W2, b2,
                                                    out, n_edges);
    } else {
        edge_direct_kernel<<<2048, 256, 0, stream>>>(h, src, dst, W1, b1, W2, b2,
                                                     out, n_edges);
    }
}